// ClauseBodyInferModule_18227841204321
// MI455X (gfx1250) — compile-verified
//
#include <hip/hip_runtime.h>
#include <hip/hip_bf16.h>
#include <stdint.h>

// Problem constants (from reference): C,G,S,L,B = 16, 8192, 8, 3, 32
#define CC 16
#define GG 8192
#define SS 8
#define LL 3
#define BB 32
#define TG 32   // g-tile per wave
#define WPB 8   // waves per block (256 threads, wave32)

typedef int v4i __attribute__((vector_size(16)));

// ---- CDNA5 async global->LDS copy (ASYNCcnt path) -------------------------
__device__ __forceinline__ void async_copy_b128(const void* g, void* l) {
#if __has_builtin(__builtin_amdgcn_global_load_async_to_lds_b128)
  typedef __attribute__((address_space(1))) v4i* gvp;   // global, 64-bit
  typedef __attribute__((address_space(3))) v4i* lvp;   // LDS, 32-bit
  gvp gp = (gvp)(unsigned long long)(uintptr_t)g;       // VA == AS1 value
  lvp lp = (lvp)(unsigned)(uintptr_t)l;                 // low 32 bits == LDS offset
  __builtin_amdgcn_global_load_async_to_lds_b128(gp, lp, /*offset*/ 0, /*cpol*/ 0);
#else
  unsigned l32 = (unsigned)(uintptr_t)l;
  asm volatile("global_load_async_to_lds_b128 %0, %1, off"
               :
               : "v"(l32), "v"(g)
               : "memory");
#endif
}

__device__ __forceinline__ void wait_async_zero() {
#if __has_builtin(__builtin_amdgcn_s_wait_asynccnt)
  __builtin_amdgcn_s_wait_asynccnt(0);
#else
  asm volatile("s_wait_asynccnt 0x0" ::: "memory");
#endif
  asm volatile("" ::: "memory");  // keep LDS reads below the wait
}

// ---- Kernel 1: gather-prod-sum. One wave = 32 batch lanes, a 32-g tile. ---
template <bool USE_XT>
__global__ __launch_bounds__(256) void clause_kernel(const float* __restrict__ xsrc,
                                                     const int* __restrict__ I,
                                                     float* __restrict__ out) {
  __shared__ __align__(16) int s_idx[WPB][TG * SS * LL];  // 3072 B per wave
  __shared__ float s_out[WPB][BB * 33];                   // 4224 B per wave

  const int wave = threadIdx.x >> 5;
  const int lane = threadIdx.x & 31;                 // lane == batch index b
  const int task = blockIdx.x * WPB + wave;          // 0 .. C*(G/TG)-1 = 4095
  const int c    = task >> 8;                        // / (G/TG) = /256
  const int g0   = (task & 255) * TG;

  // Stage this wave's 3 KB contiguous index tile into LDS asynchronously.
  const char* gI = (const char*)(I + ((size_t)c * GG + g0) * (SS * LL));
  char* lI = (char*)&s_idx[wave][0];
#pragma unroll
  for (int i = 0; i < (TG * SS * LL * 4) / 512; ++i) {   // 6 x (32 lanes * 16B)
    const int off = i * 512 + lane * 16;
    async_copy_b128(gI + off, lI + off);
  }
  wait_async_zero();

#pragma unroll 2
  for (int gi = 0; gi < TG; ++gi) {
    const int* ib = &s_idx[wave][gi * SS * LL];
    float acc = 0.0f;
#pragma unroll
    for (int s = 0; s < SS; ++s) {
      // indices are wave-uniform: move to SGPRs -> saddr-form coalesced gathers
      const int i0 = __builtin_amdgcn_readfirstlane(ib[3 * s + 0]);
      const int i1 = __builtin_amdgcn_readfirstlane(ib[3 * s + 1]);
      const int i2 = __builtin_amdgcn_readfirstlane(ib[3 * s + 2]);
      float a, b, d;
      if (USE_XT) {   // x_t(G,B): one 128B line per index for the whole wave
        a = xsrc[(size_t)i0 * BB + lane];
        b = xsrc[(size_t)i1 * BB + lane];
        d = xsrc[(size_t)i2 * BB + lane];
      } else {        // fallback: strided gather from x(B,G), L2-resident
        a = xsrc[(size_t)lane * GG + i0];
        b = xsrc[(size_t)lane * GG + i1];
        d = xsrc[(size_t)lane * GG + i2];
      }
      acc = fmaf(a * b, d, acc);
    }
    // stride-33 staging: writes hit banks (33*lane+gi) -> conflict-free
    s_out[wave][lane * 33 + gi] = acc;
  }

  // Flush: lane t reads row b=r -> coalesced 128B non-temporal stores.
  float* obase = out + (size_t)c * BB * GG + g0;
#pragma unroll 4
  for (int r = 0; r < BB; ++r) {
    __builtin_nontemporal_store(s_out[wave][r * 33 + lane],
                                obase + (size_t)r * GG + lane);
  }
}

// Force emission order: clause kernels first so the asm snippet shows the
// async global->LDS instructions.
template __global__ void clause_kernel<true>(const float* __restrict__,
                                             const int* __restrict__,
                                             float* __restrict__);
template __global__ void clause_kernel<false>(const float* __restrict__,
                                              const int* __restrict__,
                                              float* __restrict__);

// ---- Kernel 0: transpose x (B,G) -> x_t (G,B) so gathers are coalesced ----
__global__ __launch_bounds__(1024) void xpose_kernel(const float* __restrict__ x,
                                                     float* __restrict__ xt) {
  __shared__ float t[32][33];
  const int g0 = blockIdx.x * 32;
  const int tx = threadIdx.x, ty = threadIdx.y;   // blockDim = (32,32); B == 32
  t[ty][tx] = x[(size_t)ty * GG + (g0 + tx)];     // coalesced read along g
  __syncthreads();
  xt[(size_t)(g0 + ty) * BB + tx] = t[tx][ty];    // coalesced write along b
}

extern "C" void kernel_launch(void* const* d_in, const int* in_sizes, int n_in,
                              void* d_out, int out_size, void* d_ws, size_t ws_size,
                              hipStream_t stream) {
  const float* x = (const float*)d_in[0];   // (B,G) float32
  const int*   I = (const int*)d_in[1];     // (C,G,S,L) indices (int32 per harness ABI)
  float* out = (float*)d_out;               // (C,B,G) float32

  const int nBlocks = (CC * (GG / TG)) / WPB;   // 4096 wave-tasks / 8 = 512 blocks

  if (ws_size >= (size_t)GG * BB * sizeof(float)) {
    float* xt = (float*)d_ws;
    hipLaunchKernelGGL(xpose_kernel, dim3(GG / 32), dim3(32, 32), 0, stream, x, xt);
    hipLaunchKernelGGL((clause_kernel<true>), dim3(nBlocks), dim3(256), 0, stream,
                       xt, I, out);
  } else {
    hipLaunchKernelGGL((clause_kernel<false>), dim3(nBlocks), dim3(256), 0, stream,
                       x, I, out);
  }
}